// SnakeFastLoss_86827058856554
// MI455X (gfx1250) — compile-verified
//
#include <hip/hip_runtime.h>
#include <hip/hip_bf16.h>

#define HCONST 256
#define WCONST 256
#define HWC (HCONST * WCONST)
#define BATCH 4
#define NNODES 96
#define NSEG (NNODES - 1)
#define FR 6
#define KTAPS 169   // 13*13
#define KPAD 172    // padded to multiple of 4
#define NCHUNK 43   // 172 / 4
#define NSTEPS 50
#define TCOLS 140   // 128-pixel strip + 12 halo
#define TROWS 14    // 13 data rows + 1 zero row for K padding

typedef __attribute__((ext_vector_type(2))) float v2f;
typedef __attribute__((ext_vector_type(8))) float v8f;

#if __has_builtin(__builtin_amdgcn_wmma_f32_16x16x4_f32)
#define USE_WMMA_F32 1
#endif
#if __has_builtin(__builtin_amdgcn_global_load_async_to_lds_b32)
#define USE_ASYNC_LDS 1
#endif

// ---------------------------------------------------------------------------
// K1: 13x13 cross-correlation of pred and |pred| with 2 filters, 'SAME' zero
// pad, x10.  WMMA formulation: per wave, D(16x16) = A(16x172) * B(172x16)
// accumulated over 43 chunks of K=4.  A rows 0/1 hold the two filters (row 2+
// and K>=169 read an all-zero LDS region, so the loop is branch-free and EXEC
// stays all-ones).  B holds im2col patches for 16 consecutive output pixels,
// addressed through a precomputed tap-offset table (no per-iteration /13, %13).
// Block = one image row strip of 128 pixels (8 waves x 16 pixels).
// ---------------------------------------------------------------------------
__global__ __launch_bounds__(256) void conv_grad_kernel(
    const float* __restrict__ pred, const float* __restrict__ fltr,
    float* __restrict__ gv, float* __restrict__ gw)
{
    __shared__ float tile[TROWS * TCOLS]; // rows y-6..y+6 + one zero row
    __shared__ float Fl[3 * KPAD];        // fy, fx, zeros (A rows >= 2)
    __shared__ int   toff[KPAD];          // tap k -> dy*TCOLS+dx (zero row if pad)

    const int tid  = threadIdx.x;
    const int bid  = blockIdx.x;
    const int b    = bid >> 9;            // 512 blocks per batch (256 rows * 2 strips)
    const int rem  = bid & 511;
    const int y    = rem >> 1;
    const int xblk = rem & 1;
    const int xbase = xblk * 128;

    for (int e = tid; e < 3 * KPAD; e += 256) {
        int k = e % KPAD;
        Fl[e] = (e < 2 * KPAD && k < KTAPS) ? fltr[(e / KPAD) * KTAPS + k] : 0.0f;
    }
    for (int e = tid; e < KPAD; e += 256)
        toff[e] = (e < KTAPS) ? (e / 13) * TCOLS + (e % 13) : 13 * TCOLS;

    const float* predb = pred + (size_t)b * HWC;
    for (int e = tid; e < TROWS * TCOLS; e += 256) {
        int r  = e / TCOLS, cc = e % TCOLS;
        int yy = y - FR + r;
        int xx = xbase - FR + cc;
        float v = 0.0f;
        if (r < 13 && yy >= 0 && yy < HCONST && xx >= 0 && xx < WCONST)
            v = predb[yy * WCONST + xx];
        tile[e] = v;
    }
    __syncthreads();

    const int lane = tid & 31;
    const int wave = tid >> 5;
    const int x0l  = wave * 16;           // pixel offset of this wave inside strip
    const int m    = lane & 15;           // A row / B column

#ifdef USE_WMMA_F32
    v8f accv = {};                        // chain for pred   -> gv (fy,fx)
    v8f accw = {};                        // chain for |pred| -> gw (fy,fx)
    const int khalf = (lane < 16) ? 0 : 2;
    const int xm    = x0l + m;            // column base into tile
    // A row pointer: rows 0/1 = filters, anything else reads the zero row.
    const float* Arow = &Fl[((m < 2) ? m : 2) * KPAD];

    for (int c = 0; c < NCHUNK; ++c) {
        const int k0 = 4 * c + khalf;     // K of vgpr0 (vgpr1 = k0+1), per ISA layout
        v2f a;
        a[0] = Arow[k0];
        a[1] = Arow[k0 + 1];
        const float e0 = tile[toff[k0] + xm];
        const float e1 = tile[toff[k0 + 1] + xm];
        v2f bv; bv[0] = e0;        bv[1] = e1;
        v2f bw; bw[0] = fabsf(e0); bw[1] = fabsf(e1);
        accv = __builtin_amdgcn_wmma_f32_16x16x4_f32(false, a, false, bv,
                                                     (short)0, accv, false, false);
        accw = __builtin_amdgcn_wmma_f32_16x16x4_f32(false, a, false, bw,
                                                     (short)0, accw, false, false);
    }
    if (lane < 16) {                      // D rows 0/1 live in vgpr0/1 of lanes 0-15
        const int x = xbase + x0l + lane;
        const size_t o0 = ((size_t)b * 2 + 0) * HWC + (size_t)y * WCONST + x;
        const size_t o1 = ((size_t)b * 2 + 1) * HWC + (size_t)y * WCONST + x;
        gv[o0] = accv[0] * 10.0f;  gv[o1] = accv[1] * 10.0f;
        gw[o0] = accw[0] * 10.0f;  gw[o1] = accw[1] * 10.0f;
    }
#else
    // scalar fallback (also what the host pass sees)
    const int img = tid >> 7;             // 0: pred->gv, 1: |pred|->gw
    const int px  = tid & 127;
    float s0 = 0.0f, s1 = 0.0f;
    for (int k = 0; k < KTAPS; ++k) {
        float v = tile[toff[k] + px];
        if (img) v = fabsf(v);
        s0 += v * Fl[k];
        s1 += v * Fl[KPAD + k];
    }
    const int x = xbase + px;
    const size_t o0 = ((size_t)b * 2 + 0) * HWC + (size_t)y * WCONST + x;
    const size_t o1 = ((size_t)b * 2 + 1) * HWC + (size_t)y * WCONST + x;
    float* dst = img ? gw : gv;
    dst[o0] = s0 * 10.0f;  dst[o1] = s1 * 10.0f;
#endif
}

// ---------------------------------------------------------------------------
// K2: 50-step snake integration.  One block per batch sample, node i = thread i.
// ---------------------------------------------------------------------------
__device__ inline void bilin2(const float* __restrict__ img, float y, float x,
                              float& v0, float& v1)
{
    y = fminf(fmaxf(y, 0.0f), 255.0f);
    x = fminf(fmaxf(x, 0.0f), 255.0f);
    const float fy0 = floorf(y), fx0 = floorf(x);
    const int y0 = (int)fy0, x0 = (int)fx0;
    const int y1 = (y0 + 1 < 255) ? y0 + 1 : 255;
    const int x1 = (x0 + 1 < 255) ? x0 + 1 : 255;
    const float wy = y - fy0, wx = x - fx0;
    const float w00 = (1.0f - wy) * (1.0f - wx), w01 = (1.0f - wy) * wx;
    const float w10 = wy * (1.0f - wx),          w11 = wy * wx;
    const float* c0 = img;
    const float* c1 = img + HWC;
    v0 = c0[y0*256+x0]*w00 + c0[y0*256+x1]*w01 + c0[y1*256+x0]*w10 + c0[y1*256+x1]*w11;
    v1 = c1[y0*256+x0]*w00 + c1[y0*256+x1]*w01 + c1[y1*256+x0]*w10 + c1[y1*256+x1]*w11;
}

__global__ __launch_bounds__(128) void snake_kernel(
    const float* __restrict__ nodes, const float* __restrict__ widths,
    const float* __restrict__ gv, const float* __restrict__ gw,
    float* __restrict__ pfin, float* __restrict__ wfin)
{
    __shared__ float py[NNODES], px[NNODES], wd[NNODES];
    __shared__ float d2y[NNODES], d2x[NNODES];
    const int b = blockIdx.x;
    const int i = threadIdx.x;
    const bool act = (i < NNODES);
    const float* gvb = gv + (size_t)b * 2 * HWC;
    const float* gwb = gw + (size_t)b * 2 * HWC;

    if (act) {
        py[i] = nodes[(b * NNODES + i) * 2 + 0];
        px[i] = nodes[(b * NNODES + i) * 2 + 1];
        wd[i] = widths[b * NNODES + i];
    }
    __syncthreads();

    for (int s = 0; s < NSTEPS; ++s) {
        float fy = 0.0f, fx = 0.0f, ny = 0.0f, nx = 0.0f;
        if (act) {
            bilin2(gvb, py[i], px[i], fy, fx);           // external force (old p)
            const int im = (i > 0) ? i - 1 : 0;          // replicate ends
            const int ip = (i < NNODES - 1) ? i + 1 : NNODES - 1;
            ny = py[im] - 2.0f * py[i] + py[ip];
            nx = px[im] - 2.0f * px[i] + px[ip];
        }
        __syncthreads();
        if (act) { d2y[i] = ny; d2x[i] = nx; }
        __syncthreads();
        float npy = 0.0f, npx = 0.0f;
        if (act) {
            const int im = (i > 0) ? i - 1 : 0;
            const int ip = (i < NNODES - 1) ? i + 1 : NNODES - 1;
            const float d4y = d2y[im] - 2.0f * d2y[i] + d2y[ip];
            const float d4x = d2x[im] - 2.0f * d2x[i] + d2x[ip];
            npy = py[i] + 0.1f * (0.01f * d2y[i] - 0.01f * d4y + fy);
            npx = px[i] + 0.1f * (0.01f * d2x[i] - 0.01f * d4x + fx);
            npy = fminf(fmaxf(npy, 0.0f), 255.0f);
            npx = fminf(fmaxf(npx, 0.0f), 255.0f);
        }
        __syncthreads();
        if (act) {
            py[i] = npy; px[i] = npx;
            float g0, g1;
            bilin2(gwb, npy, npx, g0, g1);               // width force (new p)
            wd[i] = fmaxf(wd[i] + 0.1f * (g0 + g1), 0.5f);
        }
        __syncthreads();
    }
    if (act) {
        pfin[(b * NNODES + i) * 2 + 0] = py[i];
        pfin[(b * NNODES + i) * 2 + 1] = px[i];
        wfin[b * NNODES + i] = wd[i];
    }
}

// ---------------------------------------------------------------------------
// K3: render distance map + squared error, deterministic per-block partials.
// Segment table staged into LDS via CDNA5 async global->LDS loads.
// ---------------------------------------------------------------------------
__global__ __launch_bounds__(256) void render_loss_kernel(
    const float* __restrict__ pred, const float* __restrict__ pfin,
    const float* __restrict__ wfin, float* __restrict__ partials)
{
    __shared__ float rawP[2 * NNODES];
    __shared__ float rawW[NNODES];
    __shared__ float say[NSEG], sax[NSEG], saby[NSEG], sabx[NSEG];
    __shared__ float sinvd[NSEG], swa[NSEG], sdw[NSEG];
    __shared__ float red[256];

    const int tid = threadIdx.x;
    const int b   = blockIdx.x >> 8;                 // 256 blocks per batch
    const int pix = ((blockIdx.x & 255) << 8) | tid; // one pixel per thread
    const float* pf = pfin + b * 2 * NNODES;
    const float* wf = wfin + b * NNODES;

#ifdef USE_ASYNC_LDS
    for (int j = tid; j < 2 * NNODES; j += 256)
        __builtin_amdgcn_global_load_async_to_lds_b32(
            (int*)(pf + j), (int*)(&rawP[j]), 0, 0);
    for (int j = tid; j < NNODES; j += 256)
        __builtin_amdgcn_global_load_async_to_lds_b32(
            (int*)(wf + j), (int*)(&rawW[j]), 0, 0);
#if __has_builtin(__builtin_amdgcn_s_wait_asynccnt)
    __builtin_amdgcn_s_wait_asynccnt(0);
#else
    asm volatile("s_wait_asynccnt 0" ::: "memory");
#endif
#else
    for (int j = tid; j < 2 * NNODES; j += 256) rawP[j] = pf[j];
    for (int j = tid; j < NNODES; j += 256)     rawW[j] = wf[j];
#endif
    __syncthreads();

    if (tid < NSEG) {
        const float ay = rawP[2 * tid],     ax = rawP[2 * tid + 1];
        const float by = rawP[2 * tid + 2], bx = rawP[2 * tid + 3];
        const float aby = by - ay, abx = bx - ax;
        say[tid] = ay;  sax[tid] = ax;  saby[tid] = aby;  sabx[tid] = abx;
        sinvd[tid] = 1.0f / (aby * aby + abx * abx + 1e-8f);
        swa[tid] = rawW[tid];
        sdw[tid] = rawW[tid + 1] - rawW[tid];
    }
    __syncthreads();

    const float qy = (float)(pix >> 8);
    const float qx = (float)(pix & 255);
    float m = 3.4e38f;
    for (int s = 0; s < NSEG; ++s) {
        const float dy = qy - say[s], dx = qx - sax[s];
        float t = (dy * saby[s] + dx * sabx[s]) * sinvd[s];
        t = fminf(fmaxf(t, 0.0f), 1.0f);
        const float ey = dy - t * saby[s];
        const float ex = dx - t * sabx[s];
        const float d  = sqrtf(ey * ey + ex * ex + 1e-12f);
        const float deff = fmaxf(d - (swa[s] + t * sdw[s]), 0.0f);
        m = fminf(m, deff);
    }
    const float dm   = fminf(m, 15.0f);               // deff >= 0 already
    const float diff = pred[(size_t)b * HWC + pix] - dm;
    red[tid] = diff * diff;
    __syncthreads();
    for (int s2 = 128; s2 > 0; s2 >>= 1) {
        if (tid < s2) red[tid] += red[tid + s2];
        __syncthreads();
    }
    if (tid == 0) partials[blockIdx.x] = red[0];
}

// ---------------------------------------------------------------------------
// K4: deterministic fixed-order sum of 1024 partials -> mean loss.
// ---------------------------------------------------------------------------
__global__ __launch_bounds__(256) void finalize_kernel(
    const float* __restrict__ partials, float* __restrict__ out)
{
    __shared__ float red[256];
    const int tid = threadIdx.x;
    red[tid] = partials[tid] + partials[tid + 256] +
               partials[tid + 512] + partials[tid + 768];
    __syncthreads();
    for (int s = 128; s > 0; s >>= 1) {
        if (tid < s) red[tid] += red[tid + s];
        __syncthreads();
    }
    if (tid == 0) out[0] = red[0] * (1.0f / 262144.0f);   // mean over B*1*H*W
}

// ---------------------------------------------------------------------------
extern "C" void kernel_launch(void* const* d_in, const int* in_sizes, int n_in,
                              void* d_out, int out_size, void* d_ws, size_t ws_size,
                              hipStream_t stream)
{
    (void)in_sizes; (void)n_in; (void)out_size; (void)ws_size;
    const float* pred   = (const float*)d_in[0];   // (4,1,256,256)
    const float* nodes  = (const float*)d_in[1];   // (4,96,2)
    const float* widths = (const float*)d_in[2];   // (4,96)
    const float* fltr   = (const float*)d_in[3];   // (2,1,13,13)

    float* ws       = (float*)d_ws;
    float* gv       = ws;                                     // 4*2*65536
    float* gw       = gv + (size_t)BATCH * 2 * HWC;           // 4*2*65536
    float* pfin     = gw + (size_t)BATCH * 2 * HWC;           // 4*96*2
    float* wfin     = pfin + BATCH * 2 * NNODES;              // 4*96
    float* partials = wfin + BATCH * NNODES;                  // 1024

    conv_grad_kernel <<<2048, 256, 0, stream>>>(pred, fltr, gv, gw);
    snake_kernel     <<<BATCH, 128, 0, stream>>>(nodes, widths, gv, gw, pfin, wfin);
    render_loss_kernel<<<1024, 256, 0, stream>>>(pred, pfin, wfin, partials);
    finalize_kernel  <<<1, 256, 0, stream>>>(partials, (float*)d_out);
}